// KabschMSELoss_47528108097548
// MI455X (gfx1250) — compile-verified
//
#include <hip/hip_runtime.h>

typedef __attribute__((ext_vector_type(2))) float v2f;
typedef __attribute__((ext_vector_type(8))) float v8f;

#define BLOCK     256
#define WPB       (BLOCK / 32)   // waves per block (wave32)
#define STATS     49             // 7x7 Gram entries per block partial
#define CHUNK_PTS 128            // points per wave per pipeline step
#define CHUNK_F4  96             // float4 per array per chunk (128*3/4)

// Pass 1: streaming Gram of f(p) = (gt.x,gt.y,gt.z, 1, mo.x,mo.y,mo.z) via
// V_WMMA_F32_16X16X4_F32. Depth-2 software pipeline: chunks i and i+1 live in
// VGPRs while chunk i-1 is consumed from this wave's private LDS slice.
// No workgroup barriers in the hot loop: DS ops of a wave execute in order,
// so an intra-wave s_wait_dscnt + compiler fence is sufficient.
__global__ __launch_bounds__(BLOCK) void kabsch_gram_kernel(
    const float* __restrict__ mo, const float* __restrict__ gt,
    float* __restrict__ partials, int ngroups)
{
    __shared__ __align__(16) float stage[WPB][2 * CHUNK_PTS * 3]; // 768 f / wave
    __shared__ float red[WPB][7][16];

    const int tid  = threadIdx.x;
    const int wave = tid >> 5;
    const int lane = tid & 31;
    const int m    = lane & 15;           // A-matrix row (feature index)
    const int kb   = (lane >> 4) << 1;    // K-pair base: 0 (lanes 0-15) or 2

    // Per-lane constants selecting feature f_m from the staged points.
    const bool is_gt = (m < 3);
    const bool is_mo = (m >= 4) && (m < 7);
    const int  comp  = is_gt ? m : (is_mo ? (m - 4) : 0);
    const int  laneoff = (is_mo ? (CHUNK_PTS * 3) : 0) + kb * 3 + comp;
    const float mmask = (is_gt || is_mo) ? 1.0f : 0.0f;  // zero rows 7..15
    const float acon  = (m == 3) ? 1.0f : 0.0f;          // ones row (count/means)
    const float* Lp = &stage[wave][0];
    float4* S4 = (float4*)&stage[wave][0];

    v8f acc = {0.f, 0.f, 0.f, 0.f, 0.f, 0.f, 0.f, 0.f};

    const int tw    = (int)gridDim.x * WPB;
    const int wid   = (int)blockIdx.x * WPB + wave;
    const int niter = (ngroups + tw - 1) / tw;

    const float4* gt4 = (const float4*)gt;
    const float4* mo4 = (const float4*)mo;

    // Two in-flight register chunks (i and i+1).
    float4 a0{}, a1{}, a2{}, b0{}, b1{}, b2{};
    float4 c0{}, c1{}, c2{}, d0{}, d1{}, d2{};
    {
        int g = wid;                                    // chunk 0
        if (g < ngroups) {
            const size_t f = (size_t)g * CHUNK_F4 + (size_t)lane * 3;
            a0 = gt4[f]; a1 = gt4[f + 1]; a2 = gt4[f + 2];
            b0 = mo4[f]; b1 = mo4[f + 1]; b2 = mo4[f + 2];
        }
        g = wid + tw;                                   // chunk 1
        if (g < ngroups) {
            const size_t f = (size_t)g * CHUNK_F4 + (size_t)lane * 3;
            c0 = gt4[f]; c1 = gt4[f + 1]; c2 = gt4[f + 2];
            d0 = mo4[f]; d1 = mo4[f + 1]; d2 = mo4[f + 2];
        }
    }

    // One pipeline step: store current regs -> LDS, issue loads for chunk+2
    // into the same regs, then consume LDS with 32 chained WMMAs.
#define PIPE_STEP(IT, R0, R1, R2, R3, R4, R5)                                  \
    {                                                                          \
        const int g = wid + (IT) * tw;                                         \
        const bool active = (g < ngroups);          /* wave-uniform */         \
        if (active) {                                                          \
            S4[lane * 3 + 0] = R0;                                             \
            S4[lane * 3 + 1] = R1;                                             \
            S4[lane * 3 + 2] = R2;                                             \
            S4[CHUNK_F4 + lane * 3 + 0] = R3;                                  \
            S4[CHUNK_F4 + lane * 3 + 1] = R4;                                  \
            S4[CHUNK_F4 + lane * 3 + 2] = R5;                                  \
            asm volatile("s_wait_dscnt 0x0" ::: "memory");                     \
        }                                                                      \
        const int gn = g + 2 * tw;                                             \
        if (gn < ngroups) {                                                    \
            const size_t f = (size_t)gn * CHUNK_F4 + (size_t)lane * 3;         \
            R0 = gt4[f]; R1 = gt4[f + 1]; R2 = gt4[f + 2];                     \
            R3 = mo4[f]; R4 = mo4[f + 1]; R5 = mo4[f + 2];                     \
        }                                                                      \
        if (active) {                                                          \
            _Pragma("unroll")                                                  \
            for (int s = 0; s < 32; ++s) {                                     \
                float x0 = Lp[laneoff + 12 * s];                               \
                float x1 = Lp[laneoff + 12 * s + 3];                           \
                v2f ab = {x0 * mmask + acon, x1 * mmask + acon};               \
                acc = __builtin_amdgcn_wmma_f32_16x16x4_f32(                   \
                          false, ab, false, ab, (short)0, acc, false, false);  \
            }                                                                  \
            asm volatile("" ::: "memory");                                     \
        }                                                                      \
    }

    for (int it = 0; it < niter; it += 2) {
        PIPE_STEP(it,     a0, a1, a2, b0, b1, b2)
        PIPE_STEP(it + 1, c0, c1, c2, d0, d1, d2)
    }
#undef PIPE_STEP

    // D layout: lanes 0-15, VGPR r -> element (row r, col lane). Rows 0..6 used.
    if (lane < 16) {
#pragma unroll
        for (int r = 0; r < 7; ++r) red[wave][r][lane] = acc[r];
    }
    __syncthreads();
    if (tid < STATS) {
        const int r = tid / 7, c = tid % 7;
        float s = 0.0f;
#pragma unroll
        for (int w = 0; w < WPB; ++w) s += red[w][r][c];
        partials[(size_t)blockIdx.x * STATS + tid] = s;
    }
}

// Pass 2: deterministic fixed-order reduction of block partials (double),
// then 3x3 SVD (Jacobi on C^T C) and the analytic MSE on one thread.
__global__ void kabsch_finalize_kernel(const float* __restrict__ partials,
                                       float* __restrict__ out, int nblocks)
{
    __shared__ double G[STATS];
    const int t = threadIdx.x;
    if (t < STATS) {
        double s = 0.0;
        for (int b = 0; b < nblocks; ++b) s += (double)partials[(size_t)b * STATS + t];
        G[t] = s;
    }
    __syncthreads();
    if (t == 0) {
        const double n = G[3 * 7 + 3];               // exact point count
        double Sg[3], Sm[3], C[3][3];
        for (int a = 0; a < 3; ++a) Sg[a] = G[a * 7 + 3];
        for (int b = 0; b < 3; ++b) Sm[b] = G[3 * 7 + 4 + b];
        const double gt2 = G[0] + G[8] + G[16];
        const double mo2 = G[4 * 7 + 4] + G[5 * 7 + 5] + G[6 * 7 + 6];
        for (int a = 0; a < 3; ++a)
            for (int b = 0; b < 3; ++b)
                C[a][b] = G[a * 7 + 4 + b] - Sg[a] * Sm[b] / n;
        const double P2c = gt2 - (Sg[0]*Sg[0] + Sg[1]*Sg[1] + Sg[2]*Sg[2]) / n;
        const double T2c = mo2 - (Sm[0]*Sm[0] + Sm[1]*Sm[1] + Sm[2]*Sm[2]) / n;

        // Jacobi eigendecomposition of A = C^T C  ->  V (columns), eigenvalues
        double A[3][3], V[3][3];
        for (int i = 0; i < 3; ++i)
            for (int j = 0; j < 3; ++j) {
                double s2 = 0.0;
                for (int k = 0; k < 3; ++k) s2 += C[k][i] * C[k][j];
                A[i][j] = s2;
                V[i][j] = (i == j) ? 1.0 : 0.0;
            }
        for (int sweep = 0; sweep < 30; ++sweep) {
            for (int pi = 0; pi < 3; ++pi) {
                const int p = (pi == 2) ? 1 : 0;
                const int q = (pi == 0) ? 1 : 2;
                const double apq = A[p][q];
                if (fabs(apq) < 1e-300) continue;
                const double th = (A[q][q] - A[p][p]) / (2.0 * apq);
                const double tt = ((th >= 0.0) ? 1.0 : -1.0) /
                                  (fabs(th) + sqrt(1.0 + th * th));
                const double cs = 1.0 / sqrt(1.0 + tt * tt);
                const double sn = tt * cs;
                for (int k = 0; k < 3; ++k) {
                    const double akp = A[k][p], akq = A[k][q];
                    A[k][p] = cs * akp - sn * akq;
                    A[k][q] = sn * akp + cs * akq;
                }
                for (int k = 0; k < 3; ++k) {
                    const double apk = A[p][k], aqk = A[q][k];
                    A[p][k] = cs * apk - sn * aqk;
                    A[q][k] = sn * apk + cs * aqk;
                }
                for (int k = 0; k < 3; ++k) {
                    const double vkp = V[k][p], vkq = V[k][q];
                    V[k][p] = cs * vkp - sn * vkq;
                    V[k][q] = sn * vkp + cs * vkq;
                }
            }
        }
        // sort descending (LAPACK convention), so D=(1,1,d) hits smallest sigma
        double w[3] = {A[0][0], A[1][1], A[2][2]};
        int idx[3] = {0, 1, 2};
        for (int i = 0; i < 2; ++i)
            for (int j = i + 1; j < 3; ++j)
                if (w[idx[j]] > w[idx[i]]) { int tmp = idx[i]; idx[i] = idx[j]; idx[j] = tmp; }
        double Vs[3][3], sv[3], U[3][3];
        for (int k = 0; k < 3; ++k) {
            for (int r = 0; r < 3; ++r) Vs[r][k] = V[r][idx[k]];
            const double wk = w[idx[k]];
            sv[k] = sqrt(wk > 0.0 ? wk : 0.0);
        }
        for (int k = 0; k < 3; ++k) {
            const double inv = 1.0 / (sv[k] > 1e-300 ? sv[k] : 1e-300);
            for (int r = 0; r < 3; ++r) {
                double s2 = 0.0;
                for (int c = 0; c < 3; ++c) s2 += C[r][c] * Vs[c][k];
                U[r][k] = s2 * inv;     // u_k = C v_k / s_k
            }
        }
        auto det3 = [](const double M[3][3]) {
            return M[0][0] * (M[1][1] * M[2][2] - M[1][2] * M[2][1])
                 - M[0][1] * (M[1][0] * M[2][2] - M[1][2] * M[2][0])
                 + M[0][2] * (M[1][0] * M[2][1] - M[1][1] * M[2][0]);
        };
        const double d = (det3(U) * det3(Vs) < 0.0) ? -1.0 : 1.0;
        const double D[3] = {1.0, 1.0, d};
        // rotation = Vh @ diag(D) @ U^T ; Vh[a][k] = Vs[k][a]
        double inner = 0.0;
        for (int a = 0; a < 3; ++a)
            for (int b = 0; b < 3; ++b) {
                double rab = 0.0;
                for (int k = 0; k < 3; ++k) rab += Vs[k][a] * D[k] * U[b][k];
                inner += rab * C[a][b];
            }
        const double loss = (T2c + P2c - 2.0 * inner) / (3.0 * n);
        out[0] = (float)loss;
    }
}

extern "C" void kernel_launch(void* const* d_in, const int* in_sizes, int n_in,
                              void* d_out, int out_size, void* d_ws, size_t ws_size,
                              hipStream_t stream) {
    const float* mo = (const float*)d_in[0];   // model_output [n,3]
    const float* gt = (const float*)d_in[1];   // ground_truth [n,3] (mobile)
    float* partials = (float*)d_ws;
    const int npts = in_sizes[0] / 3;
    const int ngroups = npts / CHUNK_PTS;      // n is a multiple of 128 (4M)

    int grid = 1024;
    const size_t need = (size_t)grid * STATS * sizeof(float);
    if (ws_size < need) {
        size_t g = ws_size / (STATS * sizeof(float));
        grid = (g < 1) ? 1 : (g > 1024 ? 1024 : (int)g);
    }

    kabsch_gram_kernel<<<grid, BLOCK, 0, stream>>>(mo, gt, partials, ngroups);
    kabsch_finalize_kernel<<<1, 64, 0, stream>>>(partials, (float*)d_out, grid);
}